// DWTBlockv3_39951785787819
// MI455X (gfx1250) — compile-verified
//
#include <hip/hip_runtime.h>
#include <math.h>

typedef __attribute__((ext_vector_type(2))) float v2f;
typedef __attribute__((ext_vector_type(8))) float v8f;

#define EPSV 1e-5f

static constexpr int BN = 2;     // batch
static constexpr int C1 = 60;    // x1 channels
static constexpr int CH = 64;    // feat channels
static constexpr int R  = 4;     // reduced channels
static constexpr int D2 = 64, H2 = 64, WW = 64;
static constexpr int S  = D2 * H2 * WW;          // 262144 voxels per batch

// ---------------------------------------------------------------------------
// Kernel 1: Haar DWT over (D,H) + BN + ReLU  -> dwt[b][4][S]; raw LL -> d_out tail
// ---------------------------------------------------------------------------
__global__ void __launch_bounds__(256) k_dwt(
    const float* __restrict__ x2,
    const float* __restrict__ bn_g, const float* __restrict__ bn_b,
    const float* __restrict__ bn_m, const float* __restrict__ bn_v,
    float* __restrict__ dwt, float* __restrict__ ll_out) {
  int tid = blockIdx.x * blockDim.x + threadIdx.x;
  if (tid >= BN * S) return;
  int b = tid / S, v = tid - b * S;
  int d = v / (H2 * WW);
  int rem = v - d * (H2 * WW);
  int h = rem / WW;
  int w = rem - h * WW;
  const float* px = x2 + (size_t)b * 128 * 128 * 64;
  size_t i00 = ((size_t)(2 * d) * 128 + 2 * h) * 64 + w;
  float x00 = px[i00];
  float x01 = px[i00 + 64];             // h+1
  float x10 = px[i00 + 128 * 64];       // d+1
  float x11 = px[i00 + 128 * 64 + 64];  // d+1,h+1
  float LL = 0.5f * (x00 + x10 + x01 + x11);
  float LH = 0.5f * (x00 - x10 + x01 - x11);
  float HL = 0.5f * (x00 + x10 - x01 - x11);
  float HH = 0.5f * (x00 - x10 - x01 + x11);
  ll_out[tid] = LL;  // raw LL (pre-BN)
  float vals[4] = {LL, LH, HL, HH};
#pragma unroll
  for (int c = 0; c < 4; ++c) {
    float sc = bn_g[c] * rsqrtf(bn_v[c] + EPSV);
    float y = fmaxf((vals[c] - bn_m[c]) * sc + bn_b[c], 0.f);
    dwt[((size_t)b * 4 + c) * S + v] = y;
  }
}

// ---------------------------------------------------------------------------
// Kernel 2: 1x1x1 conv 64->4 via V_WMMA_F32_16X16X4_F32 (+BN+ReLU) and fused
// per-channel pooled sums. M=16 voxels, N=16 (4 used), K=64 as 16 x (K=4).
// ---------------------------------------------------------------------------
__global__ void __launch_bounds__(256) k_conv0_wmma(
    const float* __restrict__ dwt, const float* __restrict__ x1,
    const float* __restrict__ w0, const float* __restrict__ b0,
    const float* __restrict__ g0, const float* __restrict__ be0,
    const float* __restrict__ m0, const float* __restrict__ v0,
    float* __restrict__ s0, float* __restrict__ pooled) {
  const int lane   = threadIdx.x & 31;
  const int waveId = (blockIdx.x * blockDim.x + threadIdx.x) >> 5;
  const int nWaves = (gridDim.x * blockDim.x) >> 5;
  const int m    = lane & 15;  // A: row (voxel); B/D: column (out channel)
  const int half = lane >> 4;  // K-pair select (A/B), row+8 select (D)

  // Preload B fragments: B[k][n] = w0[n][4*kk + k], zero for n>=4.
  float bw0[16], bw1[16];
#pragma unroll
  for (int kk = 0; kk < 16; ++kk) {
    int c0 = 4 * kk + 2 * half;
    bw0[kk] = (m < R) ? w0[m * CH + c0]     : 0.f;
    bw1[kk] = (m < R) ? w0[m * CH + c0 + 1] : 0.f;
  }
  // Folded conv-bias + BN for output channel n=m (only used when m<R)
  int n4 = (m < R) ? m : 0;
  float sc = g0[n4] * rsqrtf(v0[n4] + EPSV);
  float sh = (b0[n4] - m0[n4]) * sc + be0[n4];

  const int tilesPerBatch = S / 16;
  for (int b = 0; b < BN; ++b) {
    float ps0[16], ps1[16];
#pragma unroll
    for (int kk = 0; kk < 16; ++kk) { ps0[kk] = 0.f; ps1[kk] = 0.f; }
    const float* dwtB = dwt + (size_t)b * 4 * S;
    const float* x1B  = x1  + (size_t)b * C1 * S;
    for (int t = waveId; t < tilesPerBatch; t += nWaves) {
      int vb = t * 16;
      v8f acc = {};
      {  // kk = 0 : channels 0..3 come from dwt
        int ch = 2 * half;
        float a0 = dwtB[(size_t)ch * S + vb + m];
        float a1 = dwtB[(size_t)(ch + 1) * S + vb + m];
        ps0[0] += a0; ps1[0] += a1;
        v2f A = {a0, a1};
        v2f Bf = {bw0[0], bw1[0]};
        acc = __builtin_amdgcn_wmma_f32_16x16x4_f32(
            false, A, false, Bf, (short)0, acc, false, false);
      }
#pragma unroll
      for (int kk = 1; kk < 16; ++kk) {  // channels 4..63 come from x1
        int ch = 4 * kk + 2 * half - 4;
        float a0 = x1B[(size_t)ch * S + vb + m];
        float a1 = x1B[(size_t)(ch + 1) * S + vb + m];
        ps0[kk] += a0; ps1[kk] += a1;
        v2f A = {a0, a1};
        v2f Bf = {bw0[kk], bw1[kk]};
        acc = __builtin_amdgcn_wmma_f32_16x16x4_f32(
            false, A, false, Bf, (short)0, acc, false, false);
      }
      if (m < R) {  // D: lane holds column n=m, rows i+8*half = voxel offset
        float* outp = s0 + ((size_t)b * R + m) * S + vb + 8 * half;
        float4 lo, hi;
        lo.x = fmaxf(acc[0] * sc + sh, 0.f);
        lo.y = fmaxf(acc[1] * sc + sh, 0.f);
        lo.z = fmaxf(acc[2] * sc + sh, 0.f);
        lo.w = fmaxf(acc[3] * sc + sh, 0.f);
        hi.x = fmaxf(acc[4] * sc + sh, 0.f);
        hi.y = fmaxf(acc[5] * sc + sh, 0.f);
        hi.z = fmaxf(acc[6] * sc + sh, 0.f);
        hi.w = fmaxf(acc[7] * sc + sh, 0.f);
        *(float4*)outp = lo;
        *(float4*)(outp + 4) = hi;
      }
    }
    // Reduce pooled partial sums over the 16 lanes of each half-wave.
#pragma unroll
    for (int kk = 0; kk < 16; ++kk) {
      float a = ps0[kk], c = ps1[kk];
      for (int off = 8; off >= 1; off >>= 1) {
        a += __shfl_xor(a, off, 32);
        c += __shfl_xor(c, off, 32);
      }
      if (m == 0) {
        atomicAdd(&pooled[b * CH + 4 * kk + 2 * half], a);
        atomicAdd(&pooled[b * CH + 4 * kk + 2 * half + 1], c);
      }
    }
  }
}

// ---------------------------------------------------------------------------
// Kernel 3: channel gate MLP (64->4 BN ReLU ->64), one thread per (b,c).
// ---------------------------------------------------------------------------
__global__ void k_attc(const float* __restrict__ pooled,
                       const float* __restrict__ w1, const float* __restrict__ b1,
                       const float* __restrict__ bng, const float* __restrict__ bnb,
                       const float* __restrict__ bnm, const float* __restrict__ bnv,
                       const float* __restrict__ w2, const float* __restrict__ b2,
                       float* __restrict__ att_c) {
  int tid = threadIdx.x;
  if (tid >= BN * CH) return;
  int b = tid / CH, c = tid - b * CH;
  const float invS = 1.0f / (float)S;
  float h[R];
#pragma unroll
  for (int r = 0; r < R; ++r) {
    float acc = b1[r];
    for (int cc = 0; cc < CH; ++cc)
      acc += (pooled[b * CH + cc] * invS) * w1[r * CH + cc];
    float scv = bng[r] * rsqrtf(bnv[r] + EPSV);
    h[r] = fmaxf((acc - bnm[r]) * scv + bnb[r], 0.f);
  }
  float att = b2[c];
#pragma unroll
  for (int r = 0; r < R; ++r) att += h[r] * w2[c * R + r];
  att_c[tid] = att;
}

// ---------------------------------------------------------------------------
// Kernel 4/5: dilated 3x3x3 conv (4->4, dil=4, pad=4) + BN + ReLU.
// One thread per voxel computes all 4 outputs; weights staged in LDS.
// ---------------------------------------------------------------------------
__global__ void __launch_bounds__(256) k_dilconv(
    const float* __restrict__ sin, const float* __restrict__ wgt,
    const float* __restrict__ bias,
    const float* __restrict__ g, const float* __restrict__ be,
    const float* __restrict__ mm, const float* __restrict__ vv,
    float* __restrict__ sout) {
  __shared__ float wl[R * R * 27];
  for (int i = threadIdx.x; i < R * R * 27; i += blockDim.x) wl[i] = wgt[i];
  __syncthreads();
  int tid = blockIdx.x * blockDim.x + threadIdx.x;
  if (tid >= BN * S) return;
  int b = tid / S, v = tid - b * S;
  int d = v / (H2 * WW);
  int rem = v - d * (H2 * WW);
  int h = rem / WW;
  int w = rem - h * WW;
  float acc[R];
#pragma unroll
  for (int o = 0; o < R; ++o) acc[o] = bias[o];
  const float* inB = sin + (size_t)b * R * S;
#pragma unroll
  for (int kd = 0; kd < 3; ++kd) {
    int dd = d + (kd - 1) * 4;
    if (dd < 0 || dd >= D2) continue;
#pragma unroll
    for (int kh = 0; kh < 3; ++kh) {
      int hh = h + (kh - 1) * 4;
      if (hh < 0 || hh >= H2) continue;
#pragma unroll
      for (int kw = 0; kw < 3; ++kw) {
        int ww = w + (kw - 1) * 4;
        if (ww < 0 || ww >= WW) continue;
        int vi = (dd * H2 + hh) * WW + ww;
        int kidx = kd * 9 + kh * 3 + kw;
#pragma unroll
        for (int i = 0; i < R; ++i) {
          float x = inB[(size_t)i * S + vi];
#pragma unroll
          for (int o = 0; o < R; ++o)
            acc[o] = fmaf(wl[(o * R + i) * 27 + kidx], x, acc[o]);
        }
      }
    }
  }
#pragma unroll
  for (int o = 0; o < R; ++o) {
    float scv = g[o] * rsqrtf(vv[o] + EPSV);
    float y = fmaxf((acc[o] - mm[o]) * scv + be[o], 0.f);
    sout[((size_t)b * R + o) * S + v] = y;
  }
}

// ---------------------------------------------------------------------------
// Kernel 6: fused att_s (1x1x1 conv 4->1) + sigmoid gate + feat scaling.
// feat never materialized: channels 0..3 from dwt, 4..63 from x1.
// ---------------------------------------------------------------------------
__global__ void __launch_bounds__(256) k_final(
    const float* __restrict__ dwt, const float* __restrict__ x1,
    const float* __restrict__ s2,
    const float* __restrict__ w3, const float* __restrict__ b3,
    const float* __restrict__ att_c, float* __restrict__ out) {
  __shared__ float attc_s[BN * CH];
  if (threadIdx.x < BN * CH) attc_s[threadIdx.x] = att_c[threadIdx.x];
  __syncthreads();
  int tid = blockIdx.x * blockDim.x + threadIdx.x;
  if (tid >= BN * S) return;
  int b = tid / S, v = tid - b * S;
  float atts = b3[0];
#pragma unroll
  for (int r = 0; r < R; ++r) atts += w3[r] * s2[((size_t)b * R + r) * S + v];
  const float* dwtB = dwt + (size_t)b * 4 * S;
  const float* x1B  = x1 + (size_t)b * C1 * S;
  float* outB = out + (size_t)b * CH * S;
#pragma unroll
  for (int c = 0; c < 4; ++c) {
    float z = attc_s[b * CH + c] + atts;
    float scale = 1.0f + 1.0f / (1.0f + expf(-z));
    outB[(size_t)c * S + v] = dwtB[(size_t)c * S + v] * scale;
  }
#pragma unroll 4
  for (int c = 4; c < CH; ++c) {
    float z = attc_s[b * CH + c] + atts;
    float scale = 1.0f + 1.0f / (1.0f + expf(-z));
    outB[(size_t)c * S + v] = x1B[(size_t)(c - 4) * S + v] * scale;
  }
}

// ---------------------------------------------------------------------------
extern "C" void kernel_launch(void* const* d_in, const int* in_sizes, int n_in,
                              void* d_out, int out_size, void* d_ws, size_t ws_size,
                              hipStream_t stream) {
  const float* x1     = (const float*)d_in[0];
  const float* x2     = (const float*)d_in[1];
  const float* bn_g   = (const float*)d_in[2];
  const float* bn_b   = (const float*)d_in[3];
  const float* bn_m   = (const float*)d_in[4];
  const float* bn_v   = (const float*)d_in[5];
  const float* cg_w1  = (const float*)d_in[6];
  const float* cg_b1  = (const float*)d_in[7];
  const float* cg_bng = (const float*)d_in[8];
  const float* cg_bnb = (const float*)d_in[9];
  const float* cg_bnm = (const float*)d_in[10];
  const float* cg_bnv = (const float*)d_in[11];
  const float* cg_w2  = (const float*)d_in[12];
  const float* cg_b2  = (const float*)d_in[13];
  const float* sg_w0  = (const float*)d_in[14];
  const float* sg_b0  = (const float*)d_in[15];
  const float* sg_g0  = (const float*)d_in[16];
  const float* sg_be0 = (const float*)d_in[17];
  const float* sg_m0  = (const float*)d_in[18];
  const float* sg_v0  = (const float*)d_in[19];
  const float* sg_w1  = (const float*)d_in[20];
  const float* sg_b1  = (const float*)d_in[21];
  const float* sg_g1  = (const float*)d_in[22];
  const float* sg_be1 = (const float*)d_in[23];
  const float* sg_m1  = (const float*)d_in[24];
  const float* sg_v1  = (const float*)d_in[25];
  const float* sg_w2  = (const float*)d_in[26];
  const float* sg_b2  = (const float*)d_in[27];
  const float* sg_g2  = (const float*)d_in[28];
  const float* sg_be2 = (const float*)d_in[29];
  const float* sg_m2  = (const float*)d_in[30];
  const float* sg_v2  = (const float*)d_in[31];
  const float* sg_w3  = (const float*)d_in[32];
  const float* sg_b3  = (const float*)d_in[33];

  float* ws     = (float*)d_ws;
  float* dwt    = ws;                            // B*4*S
  float* s0     = dwt + (size_t)BN * 4 * S;      // B*4*S
  float* s1     = s0  + (size_t)BN * 4 * S;      // B*4*S
  float* pooled = s1  + (size_t)BN * 4 * S;      // B*CH
  float* att_c  = pooled + BN * CH;              // B*CH
  float* s2     = s0;                            // reuse s0 (dead after conv1)

  float* out    = (float*)d_out;
  float* ll_out = out + (size_t)BN * CH * S;     // LL tail of output

  hipMemsetAsync(pooled, 0, BN * CH * sizeof(float), stream);

  int nvox = BN * S;
  k_dwt<<<(nvox + 255) / 256, 256, 0, stream>>>(x2, bn_g, bn_b, bn_m, bn_v, dwt, ll_out);
  k_conv0_wmma<<<256, 256, 0, stream>>>(dwt, x1, sg_w0, sg_b0, sg_g0, sg_be0,
                                        sg_m0, sg_v0, s0, pooled);
  k_attc<<<1, 128, 0, stream>>>(pooled, cg_w1, cg_b1, cg_bng, cg_bnb, cg_bnm,
                                cg_bnv, cg_w2, cg_b2, att_c);
  k_dilconv<<<(nvox + 255) / 256, 256, 0, stream>>>(s0, sg_w1, sg_b1, sg_g1,
                                                    sg_be1, sg_m1, sg_v1, s1);
  k_dilconv<<<(nvox + 255) / 256, 256, 0, stream>>>(s1, sg_w2, sg_b2, sg_g2,
                                                    sg_be2, sg_m2, sg_v2, s2);
  k_final<<<(nvox + 255) / 256, 256, 0, stream>>>(dwt, x1, s2, sg_w3, sg_b3,
                                                  att_c, out);
}